// IndirectGridSample_74242804678695
// MI455X (gfx1250) — compile-verified
//
#include <hip/hip_runtime.h>
#include <stdint.h>

typedef float v2f __attribute__((ext_vector_type(2)));
typedef float v4f __attribute__((ext_vector_type(4)));
typedef int   v2i_vs __attribute__((vector_size(8)));   // matches builtin param

#define GLOBAL_AS __attribute__((address_space(1)))
#define LDS_AS    __attribute__((address_space(3)))

#if __has_builtin(__builtin_amdgcn_global_load_async_to_lds_b64) && \
    __has_builtin(__builtin_amdgcn_global_load_async_to_lds_b32)
#define HAVE_ASYNC_LDS 1
#else
#define HAVE_ASYNC_LDS 0
#endif

// 4 corner gathers for one channel, shared saddr base, per-channel imm offset.
// th:TH_LOAD_HT -> bias retention: the whole 128MB input fits in the 192MB L2.
#define G4(d0,d1,d2,d3,OFF) \
  "global_load_b32 %" d0 ", %16, %20 offset:" OFF " th:TH_LOAD_HT\n\t" \
  "global_load_b32 %" d1 ", %17, %20 offset:" OFF " th:TH_LOAD_HT\n\t" \
  "global_load_b32 %" d2 ", %18, %20 offset:" OFF " th:TH_LOAD_HT\n\t" \
  "global_load_b32 %" d3 ", %19, %20 offset:" OFF " th:TH_LOAD_HT\n\t"

// 16 loads in flight (4 channels x 4 corners), then one explicit wait.
// Channel stride = H*W*4 = 262144 B, fits signed-24-bit IOFFSET (max 786432).
__device__ __forceinline__ void gather16(uint64_t base, uint32_t o00, uint32_t o01,
                                         uint32_t o10, uint32_t o11, float v[16]) {
  asm volatile(
      G4("0","1","2","3","0")
      G4("4","5","6","7","262144")
      G4("8","9","10","11","524288")
      G4("12","13","14","15","786432")
      "s_wait_loadcnt 0x0"
      : "=v"(v[0]), "=v"(v[1]), "=v"(v[2]), "=v"(v[3]),
        "=v"(v[4]), "=v"(v[5]), "=v"(v[6]), "=v"(v[7]),
        "=v"(v[8]), "=v"(v[9]), "=v"(v[10]), "=v"(v[11]),
        "=v"(v[12]), "=v"(v[13]), "=v"(v[14]), "=v"(v[15])
      : "v"(o00), "v"(o01), "v"(o10), "v"(o11), "s"(base)
      : "memory");
}

__global__ __launch_bounds__(256)
void IndirectGridSample_74242804678695_kernel(const float* __restrict__ inp,
                                              const float* __restrict__ grid,
                                              const int* __restrict__ idx,
                                              float* __restrict__ out, int P) {
  constexpr int C = 16, H = 256, W = 256;
  constexpr uint32_t HWu = 65536u;

  const int tid = threadIdx.x;
  const int p   = blockIdx.x * 256 + tid;
  const int pc  = (p < P) ? p : (P - 1);   // clamp tail lanes; stores are guarded

  float gx, gy; int ind;
#if HAVE_ASYNC_LDS
  // Stage the streaming coordinate data through LDS with async global->LDS
  // copies (ASYNCcnt path). Each lane owns its slot: no barrier needed.
  __shared__ float s_grid[512];
  __shared__ int   s_idx[256];
  __builtin_amdgcn_global_load_async_to_lds_b64(
      (GLOBAL_AS v2i_vs*)(uintptr_t)(grid + (size_t)2 * pc),
      (LDS_AS v2i_vs*)(uint32_t)(uintptr_t)(&s_grid[2 * tid]), 0, 0);
  __builtin_amdgcn_global_load_async_to_lds_b32(
      (GLOBAL_AS int*)(uintptr_t)(idx + pc),
      (LDS_AS int*)(uint32_t)(uintptr_t)(&s_idx[tid]), 0, 0);
#if __has_builtin(__builtin_amdgcn_s_wait_asynccnt)
  __builtin_amdgcn_s_wait_asynccnt(0);
#else
  asm volatile("s_wait_asynccnt 0x0" ::: "memory");
#endif
  asm volatile("" ::: "memory");
  gx  = s_grid[2 * tid];
  gy  = s_grid[2 * tid + 1];
  ind = s_idx[tid];
#else
  // Streamed once: non-temporal loads keep L2 free for the gathered input.
  v2f g = __builtin_nontemporal_load((const v2f*)grid + pc);
  gx = g.x; gy = g.y;
  ind = __builtin_nontemporal_load(idx + pc);
#endif

  if (p >= P) return;

  // Bilinear setup (matches reference semantics incl. border validity masks).
  const float x = (gx + 1.0f) * 0.5f * (float)(W - 1);
  const float y = (gy + 1.0f) * 0.5f * (float)(H - 1);
  const float x0f = floorf(x), y0f = floorf(y);
  const float wx = x - x0f, wy = y - y0f;
  const int x0 = (int)x0f, y0 = (int)y0f;
  const int x1 = x0 + 1,  y1 = y0 + 1;
  const bool vx0 = (x0 >= 0) && (x0 <= W - 1);
  const bool vx1 = (x1 >= 0) && (x1 <= W - 1);
  const bool vy0 = (y0 >= 0) && (y0 <= H - 1);
  const bool vy1 = (y1 >= 0) && (y1 <= H - 1);
  const int x0c = x0 < 0 ? 0 : (x0 > W - 1 ? W - 1 : x0);
  const int x1c = x1 < 0 ? 0 : (x1 > W - 1 ? W - 1 : x1);
  const int y0c = y0 < 0 ? 0 : (y0 > H - 1 ? H - 1 : y0);
  const int y1c = y1 < 0 ? 0 : (y1 > H - 1 ? H - 1 : y1);
  // Fold validity into the weights once; gathers use clamped (always in-bounds)
  // addresses, so no per-load predication is needed.
  const float w00 = (1.0f - wx) * (1.0f - wy) * ((vx0 && vy0) ? 1.0f : 0.0f);
  const float w01 = wx * (1.0f - wy) * ((vx1 && vy0) ? 1.0f : 0.0f);
  const float w10 = (1.0f - wx) * wy * ((vx0 && vy1) ? 1.0f : 0.0f);
  const float w11 = wx * wy * ((vx1 && vy1) ? 1.0f : 0.0f);

  // Byte offsets (max 134MB < 2^31): saddr = inp, voffset = corner, imm = channel.
  const uint32_t base = (uint32_t)ind * (uint32_t)C * HWu;
  const uint32_t r0 = base + (uint32_t)y0c * (uint32_t)W;
  const uint32_t r1 = base + (uint32_t)y1c * (uint32_t)W;
  uint32_t o00 = (r0 + (uint32_t)x0c) * 4u;
  uint32_t o01 = (r0 + (uint32_t)x1c) * 4u;
  uint32_t o10 = (r1 + (uint32_t)x0c) * 4u;
  uint32_t o11 = (r1 + (uint32_t)x1c) * 4u;

  float* outp = out + (size_t)p * 16u;
  const uint64_t ibase = (uint64_t)(uintptr_t)inp;

#pragma unroll
  for (int k = 0; k < 4; ++k) {          // 4 channels per iteration
    float v[16];
    gather16(ibase, o00, o01, o10, o11, v);
    v4f r;
    r.x = fmaf(v[3],  w11, fmaf(v[2],  w10, fmaf(v[1],  w01, v[0]  * w00)));
    r.y = fmaf(v[7],  w11, fmaf(v[6],  w10, fmaf(v[5],  w01, v[4]  * w00)));
    r.z = fmaf(v[11], w11, fmaf(v[10], w10, fmaf(v[9],  w01, v[8]  * w00)));
    r.w = fmaf(v[15], w11, fmaf(v[14], w10, fmaf(v[13], w01, v[12] * w00)));
    // 128MB output stream: b128 non-temporal store, don't pollute L2.
    __builtin_nontemporal_store(r, (v4f*)(outp + k * 4));
    o00 += 0x100000u; o01 += 0x100000u; o10 += 0x100000u; o11 += 0x100000u;
  }
}

extern "C" void kernel_launch(void* const* d_in, const int* in_sizes, int n_in,
                              void* d_out, int out_size, void* d_ws, size_t ws_size,
                              hipStream_t stream) {
  (void)n_in; (void)out_size; (void)d_ws; (void)ws_size;
  const float* inp  = (const float*)d_in[0];
  const float* grid = (const float*)d_in[1];
  const int*   idx  = (const int*)d_in[2];
  float*       out  = (float*)d_out;
  const int P = in_sizes[2];               // number of sample points
  const int blocks = (P + 255) / 256;
  hipLaunchKernelGGL(IndirectGridSample_74242804678695_kernel,
                     dim3(blocks), dim3(256), 0, stream, inp, grid, idx, out, P);
}